// GNNStream_38998303048494
// MI455X (gfx1250) — compile-verified
//
#include <hip/hip_runtime.h>
#include <hip/hip_bf16.h>

typedef __attribute__((ext_vector_type(2))) float v2f;
typedef __attribute__((ext_vector_type(8))) float v8f;

#define B_  64
#define N_  1024
#define E_  512
#define D_  12
#define BD_ 32

static __device__ __forceinline__ v2f mkv2(float a, float b) { v2f r; r.x = a; r.y = b; return r; }

// ---------------------------------------------------------------------------
// Kernel 1: h = relu(LN1(x @ we_w + we_b)) fused directly into q/k/v = h@W+b
// One thread per row (65536 rows). we_w staged in LDS (24 KB), broadcast reads.
// Memory-bound on the 134 MB read of local_feat; FMA block packed into
// v_pk_fma_f32 to stay under the HBM floor.
// ---------------------------------------------------------------------------
__global__ __launch_bounds__(256) void embed_qkv_kernel(
    const float* __restrict__ x,
    const float* __restrict__ we_w, const float* __restrict__ we_b,
    const float* __restrict__ g1,   const float* __restrict__ b1,
    const float* __restrict__ wq,   const float* __restrict__ qbias,
    const float* __restrict__ wk,   const float* __restrict__ kbias,
    const float* __restrict__ wv,   const float* __restrict__ vbias,
    float* __restrict__ qo, float* __restrict__ ko, float* __restrict__ vo)
{
    __shared__ float sW[E_ * D_];          // 24 KB
    __shared__ float sQW[D_ * D_], sKW[D_ * D_], sVW[D_ * D_];
    __shared__ float sSm[6 * D_];          // we_b, g1, b1, qb, kb, vb

    const int tid = threadIdx.x;
    for (int i = tid; i < E_ * D_; i += 256) sW[i] = we_w[i];
    if (tid < D_ * D_) { sQW[tid] = wq[tid]; sKW[tid] = wk[tid]; sVW[tid] = wv[tid]; }
    if (tid < D_) {
        sSm[0 * D_ + tid] = we_b[tid];
        sSm[1 * D_ + tid] = g1[tid];
        sSm[2 * D_ + tid] = b1[tid];
        sSm[3 * D_ + tid] = qbias[tid];
        sSm[4 * D_ + tid] = kbias[tid];
        sSm[5 * D_ + tid] = vbias[tid];
    }
    __syncthreads();

    const size_t row = (size_t)blockIdx.x * 256 + tid;   // 0..65535
    const float* __restrict__ xr = x + row * E_;

    v2f acc2[6];
    #pragma unroll
    for (int e = 0; e < 6; ++e) acc2[e] = mkv2(0.f, 0.f);

    for (int k = 0; k < E_; k += 4) {
        const float4 xv = *(const float4*)(xr + k);
        const float xs[4] = { xv.x, xv.y, xv.z, xv.w };
        #pragma unroll
        for (int j = 0; j < 4; ++j) {
            const v2f xx = mkv2(xs[j], xs[j]);
            const float* wrow = &sW[(k + j) * D_];
            const float4 w0 = *(const float4*)(wrow);
            const float4 w1 = *(const float4*)(wrow + 4);
            const float4 w2 = *(const float4*)(wrow + 8);
            acc2[0] += xx * mkv2(w0.x, w0.y);
            acc2[1] += xx * mkv2(w0.z, w0.w);
            acc2[2] += xx * mkv2(w1.x, w1.y);
            acc2[3] += xx * mkv2(w1.z, w1.w);
            acc2[4] += xx * mkv2(w2.x, w2.y);
            acc2[5] += xx * mkv2(w2.z, w2.w);
        }
    }

    float acc[D_];
    #pragma unroll
    for (int e = 0; e < 6; ++e) { acc[2 * e] = acc2[e].x; acc[2 * e + 1] = acc2[e].y; }

    // bias + LayerNorm(12) + ReLU, all in registers
    float mu = 0.f;
    #pragma unroll
    for (int d = 0; d < D_; ++d) { acc[d] += sSm[0 * D_ + d]; mu += acc[d]; }
    mu *= (1.f / D_);
    float var = 0.f;
    #pragma unroll
    for (int d = 0; d < D_; ++d) { float t = acc[d] - mu; var += t * t; }
    var *= (1.f / D_);
    const float rs = __builtin_amdgcn_rsqf(var + 1e-5f);
    float h[D_];
    #pragma unroll
    for (int d = 0; d < D_; ++d)
        h[d] = fmaxf((acc[d] - mu) * rs * sSm[1 * D_ + d] + sSm[2 * D_ + d], 0.f);

    // q/k/v projections (12x12 each)
    float q[D_], k2[D_], v2[D_];
    #pragma unroll
    for (int dd = 0; dd < D_; ++dd) {
        q[dd] = sSm[3 * D_ + dd]; k2[dd] = sSm[4 * D_ + dd]; v2[dd] = sSm[5 * D_ + dd];
    }
    #pragma unroll
    for (int d = 0; d < D_; ++d) {
        const float hv = h[d];
        #pragma unroll
        for (int dd = 0; dd < D_; ++dd) {
            q[dd]  = fmaf(hv, sQW[d * D_ + dd], q[dd]);
            k2[dd] = fmaf(hv, sKW[d * D_ + dd], k2[dd]);
            v2[dd] = fmaf(hv, sVW[d * D_ + dd], v2[dd]);
        }
    }
    float* qr = qo + row * D_;
    float* kr = ko + row * D_;
    float* vr = vo + row * D_;
    *(float4*)(qr)     = make_float4(q[0], q[1], q[2], q[3]);
    *(float4*)(qr + 4) = make_float4(q[4], q[5], q[6], q[7]);
    *(float4*)(qr + 8) = make_float4(q[8], q[9], q[10], q[11]);
    *(float4*)(kr)     = make_float4(k2[0], k2[1], k2[2], k2[3]);
    *(float4*)(kr + 4) = make_float4(k2[4], k2[5], k2[6], k2[7]);
    *(float4*)(kr + 8) = make_float4(k2[8], k2[9], k2[10], k2[11]);
    *(float4*)(vr)     = make_float4(v2[0], v2[1], v2[2], v2[3]);
    *(float4*)(vr + 4) = make_float4(v2[4], v2[5], v2[6], v2[7]);
    *(float4*)(vr + 8) = make_float4(v2[8], v2[9], v2[10], v2[11]);
}

// ---------------------------------------------------------------------------
// Kernel 2: fused attention. Block = (b, 128-row n chunk), 8 waves, wave owns
// 16 n rows. Scores computed TRANSPOSED, S'(m,n) = K * Q^T, with
// v_wmma_f32_16x16x4_f32 x3 (K=12). C-layout then puts each softmax column n
// in one lane: online softmax = in-register max/sum over 8 regs + one
// shfl_xor(16). P.V accumulated per-lane (v_pk_fma_f32) vs broadcast LDS
// reads of V rows. K/V/coords/mask staged in LDS in 4 chunks of 256 m-rows.
// ---------------------------------------------------------------------------
__global__ __launch_bounds__(256) void attn_kernel(
    const float* __restrict__ qg, const float* __restrict__ kg,
    const float* __restrict__ vg, const float* __restrict__ coords,
    const unsigned char* __restrict__ mask, float* __restrict__ hg)
{
    __shared__ float sK[256 * 14];   // stride 14: conflict-free b64 lane loads
    __shared__ float sV[256 * 16];   // stride 16: aligned b128 broadcast loads
    __shared__ float sAux[256 * 4];  // x, y, x^2+y^2, keep

    const int tid  = threadIdx.x;
    const int lane = tid & 31;
    const int wave = tid >> 5;
    const int hi   = lane >> 4;      // half-wave
    const int ln   = lane & 15;
    const int b    = blockIdx.y;
    const int n    = blockIdx.x * 128 + wave * 16 + ln;
    const size_t nb = (size_t)b * N_ + n;

    // Q tile in B-layout (4x16 per K-chunk), pre-scaled by 1/sqrt(12)
    const float s12 = 0.28867513459481288f;
    v2f qb0, qb1, qb2;
    {
        const float* qr = qg + nb * D_ + 2 * hi;
        qb0 = *(const v2f*)(qr);     qb0 *= s12;
        qb1 = *(const v2f*)(qr + 4); qb1 *= s12;
        qb2 = *(const v2f*)(qr + 8); qb2 *= s12;
    }
    const float xn  = coords[nb * 2 + 0];
    const float yn  = coords[nb * 2 + 1];
    const float sqn = xn * xn + yn * yn;

    v2f acc2[6];
    #pragma unroll
    for (int e = 0; e < 6; ++e) acc2[e] = mkv2(0.f, 0.f);
    float denom = 0.f, run = -1e30f;

    for (int ch = 0; ch < 4; ++ch) {
        { // cooperative stage: one m-row per thread
            const size_t mb = (size_t)b * N_ + ch * 256 + tid;
            const float* kr = kg + mb * D_;
            const float* vr = vg + mb * D_;
            #pragma unroll
            for (int i = 0; i < D_; i += 2)           // 6x ds_store_b64 (8B aligned)
                *(v2f*)&sK[tid * 14 + i] = *(const v2f*)(kr + i);
            *(float4*)&sV[tid * 16 + 0] = *(const float4*)(vr);
            *(float4*)&sV[tid * 16 + 4] = *(const float4*)(vr + 4);
            *(float4*)&sV[tid * 16 + 8] = *(const float4*)(vr + 8);
            const float mx = coords[mb * 2 + 0], my = coords[mb * 2 + 1];
            *(float4*)&sAux[tid * 4] =
                make_float4(mx, my, mx * mx + my * my, mask[mb] ? 0.f : 1.f);
        }
        __syncthreads();

        for (int mt = 0; mt < 16; ++mt) {
            // A tile (16m x 4k chunks) from LDS, per-lane (stride 14 -> no conflicts)
            const float* ka = &sK[(mt * 16 + ln) * 14 + 2 * hi];
            const v2f a0 = *(const v2f*)(ka);
            const v2f a1 = *(const v2f*)(ka + 4);
            const v2f a2 = *(const v2f*)(ka + 8);
            v8f c = {};
            c = __builtin_amdgcn_wmma_f32_16x16x4_f32(false, a0, false, qb0, (short)0, c, false, false);
            c = __builtin_amdgcn_wmma_f32_16x16x4_f32(false, a1, false, qb1, (short)0, c, false, false);
            c = __builtin_amdgcn_wmma_f32_16x16x4_f32(false, a2, false, qb2, (short)0, c, false, false);
            // c[r], this lane: m = mt*16 + r + 8*hi, n = this lane's n

            float ad[8];
            #pragma unroll
            for (int r = 0; r < 8; ++r) {
                const float4 aux = *(const float4*)&sAux[(mt * 16 + r + 8 * hi) * 4];
                const float d2 = fmaxf(sqn + aux.z - 2.f * (xn * aux.x + yn * aux.y), 0.f);
                ad[r] = __expf(-5.f * __builtin_amdgcn_sqrtf(d2)) * aux.w;
            }

            // online softmax over this tile's 16 m values of column n
            float tmax = fmaxf(c[0], c[1]);
            #pragma unroll
            for (int r = 2; r < 8; ++r) tmax = fmaxf(tmax, c[r]);
            tmax = fmaxf(tmax, __shfl_xor(tmax, 16, 32));
            const float nm = fmaxf(run, tmax);
            const float corr = __expf(run - nm);
            run = nm;
            denom *= corr;
            const v2f corr2 = mkv2(corr, corr);
            #pragma unroll
            for (int e = 0; e < 6; ++e) acc2[e] *= corr2;

            #pragma unroll
            for (int r = 0; r < 8; ++r) {
                const float e0 = __expf(c[r] - nm);
                denom += e0;                   // softmax denom: ALL m (mask after softmax)
                const float p = e0 * ad[r];    // numerator weight
                const v2f p2 = mkv2(p, p);
                const float* vp = &sV[(mt * 16 + r + 8 * hi) * 16];
                const float4 v0 = *(const float4*)(vp);
                const float4 v1 = *(const float4*)(vp + 4);
                const float4 v2 = *(const float4*)(vp + 8);
                acc2[0] += p2 * mkv2(v0.x, v0.y);
                acc2[1] += p2 * mkv2(v0.z, v0.w);
                acc2[2] += p2 * mkv2(v1.x, v1.y);
                acc2[3] += p2 * mkv2(v1.z, v1.w);
                acc2[4] += p2 * mkv2(v2.x, v2.y);
                acc2[5] += p2 * mkv2(v2.z, v2.w);
            }
        }
        __syncthreads();
    }

    // combine half-waves (m 0..7 part + m 8..15 part of each tile)
    float acc[D_];
    #pragma unroll
    for (int e = 0; e < 6; ++e) { acc[2 * e] = acc2[e].x; acc[2 * e + 1] = acc2[e].y; }
    #pragma unroll
    for (int d = 0; d < D_; ++d) acc[d] += __shfl_xor(acc[d], 16, 32);
    denom += __shfl_xor(denom, 16, 32);

    if (hi == 0) {
        const float inv = 1.f / denom;
        float* o = hg + nb * D_;
        #pragma unroll
        for (int d = 0; d < D_; ++d) o[d] = acc[d] * inv;
    }
}

// ---------------------------------------------------------------------------
// Kernel 3: out = relu(LN2(h_graph @ bn_w + bn_b)); masked mean over N.
// One block per b; deterministic two-stage LDS reduction (no float atomics).
// ---------------------------------------------------------------------------
__global__ __launch_bounds__(256) void pool_kernel(
    const float* __restrict__ hg, const unsigned char* __restrict__ mask,
    const float* __restrict__ bw, const float* __restrict__ bb,
    const float* __restrict__ g2, const float* __restrict__ b2,
    float* __restrict__ out)
{
    __shared__ float sW[D_ * BD_];
    __shared__ float sBB[BD_], sG[BD_], sB2[BD_];
    __shared__ float psh[BD_ * 256];   // 32 KB partials, [dd][t]
    __shared__ float csh[256];

    const int tid = threadIdx.x;
    const int b = blockIdx.x;
    if (tid < D_ * BD_) sW[tid] = bw[tid];
    if (tid < BD_) { sBB[tid] = bb[tid]; sG[tid] = g2[tid]; sB2[tid] = b2[tid]; }
    __syncthreads();

    float ps[BD_];
    #pragma unroll
    for (int dd = 0; dd < BD_; ++dd) ps[dd] = 0.f;
    float cnt = 0.f;

    for (int n = tid; n < N_; n += 256) {
        const size_t nb = (size_t)b * N_ + n;
        if (!mask[nb]) {
            const float* hr = hg + nb * D_;
            const float4 h0 = *(const float4*)(hr);
            const float4 h1 = *(const float4*)(hr + 4);
            const float4 h2 = *(const float4*)(hr + 8);
            const float h[D_] = { h0.x, h0.y, h0.z, h0.w, h1.x, h1.y, h1.z, h1.w,
                                  h2.x, h2.y, h2.z, h2.w };
            float gg[BD_];
            #pragma unroll
            for (int dd = 0; dd < BD_; ++dd) gg[dd] = sBB[dd];
            #pragma unroll
            for (int d = 0; d < D_; ++d) {
                const float hv = h[d];
                #pragma unroll
                for (int dd = 0; dd < BD_; ++dd) gg[dd] = fmaf(hv, sW[d * BD_ + dd], gg[dd]);
            }
            float mu = 0.f;
            #pragma unroll
            for (int dd = 0; dd < BD_; ++dd) mu += gg[dd];
            mu *= (1.f / BD_);
            float var = 0.f;
            #pragma unroll
            for (int dd = 0; dd < BD_; ++dd) { float t = gg[dd] - mu; var += t * t; }
            var *= (1.f / BD_);
            const float rs = __builtin_amdgcn_rsqf(var + 1e-5f);
            #pragma unroll
            for (int dd = 0; dd < BD_; ++dd)
                ps[dd] += fmaxf((gg[dd] - mu) * rs * sG[dd] + sB2[dd], 0.f);
            cnt += 1.f;
        }
    }

    #pragma unroll
    for (int dd = 0; dd < BD_; ++dd) psh[dd * 256 + tid] = ps[dd];
    csh[tid] = cnt;
    __syncthreads();

    if (tid < BD_) {
        float s = 0.f, c = 0.f;
        for (int t = 0; t < 256; ++t) { s += psh[tid * 256 + t]; c += csh[t]; }
        out[(size_t)b * BD_ + tid] = s / fmaxf(c, 1e-9f);
    }
}

// ---------------------------------------------------------------------------
extern "C" void kernel_launch(void* const* d_in, const int* in_sizes, int n_in,
                              void* d_out, int out_size, void* d_ws, size_t ws_size,
                              hipStream_t stream)
{
    const float* local_feat = (const float*)d_in[0];
    const float* coords     = (const float*)d_in[1];
    const unsigned char* mask = (const unsigned char*)d_in[2];  // jnp bool_: 1 byte
    const float* we_w = (const float*)d_in[3];
    const float* we_b = (const float*)d_in[4];
    const float* ln1g = (const float*)d_in[5];
    const float* ln1b = (const float*)d_in[6];
    const float* wq_w = (const float*)d_in[7];
    const float* wq_b = (const float*)d_in[8];
    const float* wk_w = (const float*)d_in[9];
    const float* wk_b = (const float*)d_in[10];
    const float* wv_w = (const float*)d_in[11];
    const float* wv_b = (const float*)d_in[12];
    const float* bn_w = (const float*)d_in[13];
    const float* bn_b = (const float*)d_in[14];
    const float* ln2g = (const float*)d_in[15];
    const float* ln2b = (const float*)d_in[16];

    const size_t rows = (size_t)B_ * N_;          // 65536
    float* ws = (float*)d_ws;
    float* q  = ws;
    float* k  = q  + rows * D_;
    float* v  = k  + rows * D_;
    float* hg = v  + rows * D_;                   // total 12.6 MB < ws_size

    embed_qkv_kernel<<<256, 256, 0, stream>>>(local_feat, we_w, we_b, ln1g, ln1b,
                                              wq_w, wq_b, wk_w, wk_b, wv_w, wv_b,
                                              q, k, v);
    attn_kernel<<<dim3(8, B_), 256, 0, stream>>>(q, k, v, coords, mask, hg);
    pool_kernel<<<B_, 256, 0, stream>>>(hg, mask, bn_w, bn_b, ln2g, ln2b,
                                        (float*)d_out);
}